// StochasticAdvectionKernel_54271206752295
// MI455X (gfx1250) — compile-verified
//
#include <hip/hip_runtime.h>
#include <hip/hip_bf16.h>
#include <math.h>

#define CLIPV 1.0e4f
#define DTC 0.1f
#define JIT 1.0e-5f
#define MIN_SCALE 1.0e-4f

typedef float v2f __attribute__((ext_vector_type(2)));
typedef float v4f __attribute__((ext_vector_type(4)));
typedef float v8f __attribute__((ext_vector_type(8)));

__device__ __forceinline__ float san(float v) {
    if (isnan(v)) return 0.0f;
    return fminf(fmaxf(v, -CLIPV), CLIPV);
}

// ---------------------------------------------------------------------------
// Phase 1: per-(t,i,j) 2x2 PD repair + inverse + logdet + drift + scale.
// 128 items total -> 1 block of 128 threads. Writes drift/repaired outputs
// and an 8-float param record per item into d_ws.
// ---------------------------------------------------------------------------
__global__ __launch_bounds__(128)
void sa_precompute_kernel(const float* __restrict__ means,
                          const float* __restrict__ covs,
                          const float* __restrict__ sraw,
                          float* __restrict__ drift_out,
                          float* __restrict__ rep_out,
                          float* __restrict__ params)
{
    int id = threadIdx.x;
    if (id >= 128) return;
    int j = id & 1, i = (id >> 1) & 1, t = id >> 2;

    // sanitized means, drift
    float mi0 = san(means[(t*2+i)*2+0]);
    float mi1 = san(means[(t*2+i)*2+1]);
    float mj0 = san(means[(t*2+j)*2+0]);
    float mj1 = san(means[(t*2+j)*2+1]);
    float dx, dy;
    if (i == j) { dx = DTC*mi0; dy = DTC*mi1; }
    else        { dx = DTC*(mi0-mj0); dy = DTC*(mi1-mj1); }
    dx = san(dx); dy = san(dy);

    // sanitized + symmetrized covariances, rel
    const float* Ci = covs + (size_t)(t*2+i)*4;
    const float* Cj = covs + (size_t)(t*2+j)*4;
    float ci00 = san(Ci[0]), ci01 = 0.5f*(san(Ci[1])+san(Ci[2])), ci11 = san(Ci[3]);
    float cj00 = san(Cj[0]), cj01 = 0.5f*(san(Cj[1])+san(Cj[2])), cj11 = san(Cj[3]);
    float r00, r01, r11;
    if (i == j) { r00 = ci00; r01 = ci01; r11 = ci11; }
    else        { r00 = ci00+cj00; r01 = ci01+cj01; r11 = ci11+cj11; }

    // disp = I + 2*DT^2*rel + JITTER*I  (2*DT^2 = 0.02)
    float a  = san(1.0f + 0.02f*r00 + JIT);
    float bb = san(0.02f*r01);
    float c  = san(1.0f + 0.02f*r11 + JIT);

    float dsc = fmaxf(0.5f*(fabsf(a)+fabsf(c)), 1.0f);
    const float j0 = 1.0e-5f;  // max(JITTER, 1e-6)

    bool found = false;
    float aa = 0.f, cc = 0.f, ldet = 0.f, i00 = 0.f, i01 = 0.f, i11 = 0.f;
    float jit = j0;
    for (int k = 0; k < 8; ++k) {
        float at = san(a + jit*dsc);
        float ct = san(c + jit*dsc);
        if (!found && at > 0.0f) {
            float rem = ct - bb*bb/at;   // Cholesky l11^2
            if (rem > 0.0f) {
                found = true;
                aa = at; cc = ct;
                float det = at * rem;
                ldet = logf(det);
                float idet = 1.0f / det;
                i00 = san( ct*idet); i01 = san(-bb*idet); i11 = san( at*idet);
            }
        }
        jit *= 10.0f;
    }
    float rep00, rep01, rep11;
    if (found) { rep00 = aa; rep01 = bb; rep11 = cc; }
    else {
        float jfin = j0 * 1.0e8f;       // after MAX_TRIES escalations
        float add  = jfin * dsc;
        float sd0 = fmaxf(fabsf(a), add + 1.0e-4f);
        float sd1 = fmaxf(fabsf(c), add + 1.0e-4f);
        rep00 = san(sd0 + add); rep11 = san(sd1 + add); rep01 = 0.0f;
        float d0 = fmaxf(rep00, 1.0e-6f), d1 = fmaxf(rep11, 1.0e-6f);
        i00 = san(1.0f/d0); i01 = 0.0f; i11 = san(1.0f/d1);
        ldet = logf(d0) + logf(d1);
    }
    if (isnan(ldet)) ldet = 0.0f;
    if (isinf(ldet)) ldet = (ldet > 0.0f) ? 20.0f : -20.0f;

    float raw = sraw[i*2 + j];
    float sp  = (raw > 20.0f) ? raw : log1pf(expf(raw));
    float scale = sp + MIN_SCALE;

    int base = (t*2 + i)*2 + j;
    drift_out[base*2 + 0] = dx;
    drift_out[base*2 + 1] = dy;
    rep_out[base*4 + 0] = rep00;
    rep_out[base*4 + 1] = rep01;
    rep_out[base*4 + 2] = rep01;
    rep_out[base*4 + 3] = rep11;
    float* P = params + base*8;
    P[0] = i00; P[1] = i01; P[2] = i11; P[3] = 0.5f*ldet;
    P[4] = dx;  P[5] = dy;  P[6] = scale; P[7] = 0.0f;
}

// ---------------------------------------------------------------------------
// Phase 2: fused tile compute (WMMA 16x16x4 F32) + exp + triple row
// normalization + identity mix, one global write pass (b128 vectorized).
// grid = (t=32, i=2, atile=32); block = 128 threads (4 waves).
// quad[a,b] = p_a . v_b + qa_a + qb_b  as A(16x4) x B(4x16), K=4 exact.
// ---------------------------------------------------------------------------
__global__ __launch_bounds__(128)
void sa_trans_kernel(const float* __restrict__ site,
                     const float* __restrict__ params,
                     const float* __restrict__ mix_logit,
                     float* __restrict__ out)
{
    __shared__ __align__(16) float s_tile[16 * 1024];   // 64 KB row stripe

    const int t     = blockIdx.x;
    const int i     = blockIdx.y;
    const int atile = blockIdx.z;
    const int tid   = threadIdx.x;
    const int lane  = tid & 31;
    const int wave  = tid >> 5;
    const int idx   = lane & 15;   // M / N within tile
    const int half  = lane >> 4;   // K-half selector for A/B frags

    const v2f* site2 = (const v2f*)site;

    // row side (a) coordinates: fixed for the whole workgroup
    const int a = atile*16 + idx;
    const v2f ua = site2[a];
    const float ux = san(ua.x);
    const float uy = san(ua.y);

    // per column-component jj: one A fragment, 8 b-tiles per wave
    for (int jj = 0; jj < 2; ++jj) {
        const float* P = params + (size_t)(((t*2 + i)*2) + jj) * 8;
        const float m00 = P[0], m01 = P[1], m11 = P[2], halfld = P[3];
        const float ddx = P[4], ddy = P[5], scale = P[6];

        // A row a: [p0, p1, qa, 1] with p = -2*M*u, qa = u^T M u
        const float t0 = m00*ux + m01*uy;
        const float t1 = m01*ux + m11*uy;
        const float qa = ux*t0 + uy*t1;
        v2f Af;
        Af.x = half ? qa   : (-2.0f*t0);   // K=0 | K=2
        Af.y = half ? 1.0f : (-2.0f*t1);   // K=1 | K=3

        for (int btile = wave; btile < 32; btile += 4) {
            // B col b: [v0, v1, 1, qb] with v = s_b + d, qb = v^T M v
            const int b = btile*16 + idx;
            const v2f vb = site2[b];
            const float vx = san(vb.x) + ddx;
            const float vy = san(vb.y) + ddy;
            const float s0 = m00*vx + m01*vy;
            const float s1 = m01*vx + m11*vy;
            const float qb = vx*s0 + vy*s1;
            v2f Bf;
            Bf.x = half ? 1.0f : vx;           // K=0 | K=2
            Bf.y = half ? qb   : vy;           // K=1 | K=3

            v8f Cz = {0.f, 0.f, 0.f, 0.f, 0.f, 0.f, 0.f, 0.f};
            v8f D = __builtin_amdgcn_wmma_f32_16x16x4_f32(
                false, Af, false, Bf, (short)0, Cz, false, false);

            const int colbase = (jj << 9) + (btile << 4) + idx;
            const int rowbase = half * 8;
            #pragma unroll
            for (int v = 0; v < 8; ++v) {
                float qv = D[v];
                if (isnan(qv)) qv = 0.0f;
                qv = fminf(fmaxf(qv, 0.0f), 60.0f);
                float e = fminf(fmaxf(-halfld - qv, -60.0f), 20.0f);
                s_tile[(rowbase + v)*1024 + colbase] = scale * __expf(e);
            }
        }
    }
    __syncthreads();

    // Row sums: wave w owns local rows 4w..4w+3; 8 lanes x 128 cols per row,
    // vectorized as 32 x ds_load_b128 per lane.
    const float mix = 1.0f / (1.0f + __expf(-mix_logit[0]));
    const int rloc = wave*4 + (lane >> 3);
    const int seg  = lane & 7;
    float partial = 0.0f;
    const v4f* rowp = (const v4f*)(s_tile + rloc*1024 + seg*128);
    #pragma unroll 8
    for (int cidx = 0; cidx < 32; ++cidx) {
        v4f vv = rowp[cidx];
        partial += (vv.x + vv.y) + (vv.z + vv.w);
    }
    partial += __shfl_xor(partial, 1);
    partial += __shfl_xor(partial, 2);
    partial += __shfl_xor(partial, 4);

    // Fold:  t1 = k/max(s0,J); t2 = t1/max(s1,J); t3 = (1-mix)t2 + mix*I;
    //        out = t3/max(s3,J)  ==>  out = f*k + g*delta
    const float s0m  = fmaxf(partial, JIT);
    const float s1v  = partial / s0m;
    const float s1m  = fmaxf(s1v, JIT);
    const float sum2 = s1v / s1m;
    const float sum3 = (1.0f - mix)*sum2 + mix;
    const float s3m  = fmaxf(sum3, JIT);
    const float fnorm = (1.0f - mix) / (s0m * s1m * s3m);
    const float gdiag = mix / s3m;

    // Write pass: wave w writes its own rows, b128-vectorized & coalesced.
    const int Rbase = i*512 + atile*16;
    float* outt = out + (size_t)t * (1024u*1024u);
    for (int rl = 0; rl < 4; ++rl) {
        const int lr = wave*4 + rl;
        const float fr = __shfl(fnorm, rl*8);
        const float gr = __shfl(gdiag, rl*8);
        const int R = Rbase + lr;
        v4f* orow = (v4f*)(outt + (size_t)R * 1024);
        const v4f* srow = (const v4f*)(s_tile + lr*1024);
        #pragma unroll 2
        for (int c4 = lane; c4 < 256; c4 += 32) {
            v4f vs = srow[c4];
            const int cb = c4 << 2;
            v4f vo;
            vo.x = vs.x*fr + ((cb + 0 == R) ? gr : 0.0f);
            vo.y = vs.y*fr + ((cb + 1 == R) ? gr : 0.0f);
            vo.z = vs.z*fr + ((cb + 2 == R) ? gr : 0.0f);
            vo.w = vs.w*fr + ((cb + 3 == R) ? gr : 0.0f);
            orow[c4] = vo;
        }
    }
}

// ---------------------------------------------------------------------------
// Launch. Outputs concatenated: trans [32,1024,1024], drift [32,2,2,2],
// repaired [32,2,2,2,2].
// ---------------------------------------------------------------------------
extern "C" void kernel_launch(void* const* d_in, const int* in_sizes, int n_in,
                              void* d_out, int out_size, void* d_ws, size_t ws_size,
                              hipStream_t stream) {
    const float* means = (const float*)d_in[0];
    const float* covs  = (const float*)d_in[1];
    const float* site  = (const float*)d_in[2];
    const float* sraw  = (const float*)d_in[3];
    const float* mixl  = (const float*)d_in[4];

    float* out       = (float*)d_out;
    float* drift_out = out + (size_t)32*1024*1024;        // 33554432
    float* rep_out   = drift_out + 256;
    float* params    = (float*)d_ws;                      // 128 * 8 floats = 4 KB

    sa_precompute_kernel<<<1, 128, 0, stream>>>(means, covs, sraw,
                                                drift_out, rep_out, params);

    dim3 grid(32, 2, 32);   // t, i, atile
    sa_trans_kernel<<<grid, 128, 0, stream>>>(site, params, mixl, out);
}